// GAT_69483980914696
// MI455X (gfx1250) — compile-verified
//
#include <hip/hip_runtime.h>
#include <hip/hip_bf16.h>
#include <math.h>

// ---------------------------------------------------------------------------
// GAT (2-layer, PyG-style) for MI455X / gfx1250, wave32.
//   Layer: XH = X @ W (WMMA f32 16x16x4, W staged into LDS via TDM) ;
//          a_src/a_dst = <xh, att> ;
//          segment-softmax over edges by dst (3 atomic passes) ;
//          agg[dst] += alpha * xh[src]  (wave-per-edge, float4 per lane)
// ---------------------------------------------------------------------------

typedef __attribute__((ext_vector_type(2))) float        v2f;
typedef __attribute__((ext_vector_type(8))) float        v8f;
typedef __attribute__((ext_vector_type(4))) unsigned int v4u;
typedef __attribute__((ext_vector_type(8))) int          v8i;
typedef __attribute__((ext_vector_type(4))) int          v4i;

constexpr int FDIM = 128;   // F_in = H*C = 128 for both layers
constexpr int HEADS = 4;
constexpr float NEG_SLOPE = 0.2f;

// ---------------------------------------------------------------------------
// TDM: DMA a 2D f32 tile (tile_d0 elems/row x tile_d1 rows, row stride
// stride0 elems) from global memory into LDS. Descriptor layout per CDNA5
// ISA ch.8 (D# groups). Issued by one wave; completion via TENSORcnt.
// ---------------------------------------------------------------------------
__device__ __forceinline__ void tdm_load_2d_f32(unsigned lds_addr, const void* gptr,
                                                unsigned tile_d0, unsigned tile_d1,
                                                unsigned tens_d0, unsigned tens_d1,
                                                unsigned stride0) {
  unsigned long long ga = (unsigned long long)(uintptr_t)gptr;
  v4u g0;
  g0.x = 1u;                                            // count=1 (valid), user mode
  g0.y = lds_addr;                                      // LDS dest byte address
  g0.z = (unsigned)(ga & 0xFFFFFFFFu);                  // global_addr[31:0]
  g0.w = (unsigned)((ga >> 32) & 0x1FFFFFFu)            // global_addr[56:32]
       | (2u << 30);                                    // type=2 ("image")
  v8i g1;
  g1[0] = (int)(2u << 16);                              // wg_mask=0, data_size=2 (4B)
  g1[1] = (int)((tens_d0 & 0xFFFFu) << 16);             // tensor_dim0[15:0]
  g1[2] = (int)(((tens_d0 >> 16) & 0xFFFFu) | ((tens_d1 & 0xFFFFu) << 16));
  g1[3] = (int)(((tens_d1 >> 16) & 0xFFFFu) | ((tile_d0 & 0xFFFFu) << 16));
  g1[4] = (int)(tile_d1 & 0xFFFFu);                     // tile_dim1, tile_dim2=0
  g1[5] = (int)stride0;                                 // tensor_dim0_stride[31:0]
  g1[6] = 0;
  g1[7] = 0;
  v4i g2 = {0, 0, 0, 0};
  v4i g3 = {0, 0, 0, 0};
#if defined(__clang_major__) && (__clang_major__ >= 23)
  v8i g4 = {0, 0, 0, 0, 0, 0, 0, 0};
  __builtin_amdgcn_tensor_load_to_lds(g0, g1, g2, g3, g4, 0);
#else
  __builtin_amdgcn_tensor_load_to_lds(g0, g1, g2, g3, 0);
#endif
}

// ---------------------------------------------------------------------------
// GEMM: XH[nrows x 128] = X[nrows x 128] @ W[128 x 128]
// Block = 256 threads = 8 waves. Wave 0 DMAs all of W (64KB) into LDS via
// TDM; the block cooperatively stages its 16-row X panel (padded). Wave w
// computes the 16x16 output tile at columns [16w, 16w+16) with 32 straight-
// line V_WMMA_F32_16X16X4_F32, all operands fed from LDS.
// A layout (16x4 f32): lanes 0-15 -> M=lane, {K=k0,k0+1}; lanes 16-31 ->
// M=lane-16, {K=k0+2,k0+3}.  B (4x16) mirrored; C/D: VGPR r holds row r
// (lanes 0-15) / row r+8 (lanes 16-31), N = lane%16.
// ---------------------------------------------------------------------------
__global__ void __launch_bounds__(256)
gat_gemm128_wmma(const float* __restrict__ X, const float* __restrict__ W,
                 float* __restrict__ XH) {
  __shared__ float lds_w[FDIM][FDIM];      // 64KB, filled by TDM, row-major
  __shared__ float lds_x[16][FDIM + 4];    // +4 pad: conflict-free A-frag reads
  const int tid  = threadIdx.x;
  const long row0 = (long)blockIdx.x * 16;

  // Wave 0: one TDM descriptor moves the whole 128x128 W into LDS.
  if (tid < 32) {
    tdm_load_2d_f32((unsigned)(uintptr_t)&lds_w[0][0], W,
                    FDIM, FDIM, FDIM, FDIM, FDIM);
  }

  // All waves: cooperative stage of the 16x128 X panel (512 float4, 2/thread).
  {
    const float4* __restrict__ xp = (const float4*)(X + row0 * FDIM);
    #pragma unroll
    for (int i = 0; i < 2; ++i) {
      int idx = tid + i * 256;            // 0..511
      int r   = idx >> 5;                 // row 0..15
      int c4  = idx & 31;                 // float4 column
      *(float4*)&lds_x[r][c4 * 4] = xp[(long)r * (FDIM / 4) + c4];
    }
  }
  if (tid < 32) __builtin_amdgcn_s_wait_tensorcnt(0);   // wave 0's DMA done
  __syncthreads();                                      // publish LDS to all waves

  const int wave  = tid >> 5;
  const int lane  = tid & 31;
  const int col0  = wave * 16;
  const int mrow  = lane & 15;
  const int khalf = (lane >> 4) << 1;     // 0 or 2

  v8f acc = {};
  #pragma unroll
  for (int k0 = 0; k0 < FDIM; k0 += 4) {
    v2f a, b;
    a.x = lds_x[mrow][k0 + khalf];
    a.y = lds_x[mrow][k0 + khalf + 1];
    b.x = lds_w[k0 + khalf][col0 + mrow];
    b.y = lds_w[k0 + khalf + 1][col0 + mrow];
    // (neg_a, A, neg_b, B, c_mod, C, reuse_a, reuse_b)
    acc = __builtin_amdgcn_wmma_f32_16x16x4_f32(false, a, false, b,
                                                (short)0, acc, false, false);
  }

  const int rbase = (lane >> 4) << 3;     // 0 or 8
  float* outp = XH + (row0 + rbase) * FDIM + col0 + mrow;
  #pragma unroll
  for (int r = 0; r < 8; ++r)
    outp[(long)r * FDIM] = acc[r];
}

// ---------------------------------------------------------------------------
// Per-(node,head) attention logits: a_src[n,h] = <xh[n,h,:], att_src[h,:]>
// ---------------------------------------------------------------------------
__global__ void __launch_bounds__(256)
gat_att_logits(const float* __restrict__ xh,
               const float* __restrict__ att_src, const float* __restrict__ att_dst,
               float* __restrict__ a_src, float* __restrict__ a_dst, int n) {
  int idx = blockIdx.x * blockDim.x + threadIdx.x;   // (node, head) flat
  if (idx >= n * HEADS) return;
  int node = idx >> 2, h = idx & 3;
  const float4* xp = (const float4*)(xh + (long)node * FDIM + h * 32);
  const float4* sp = (const float4*)(att_src + h * 32);
  const float4* dp = (const float4*)(att_dst + h * 32);
  float ss = 0.f, ds = 0.f;
  #pragma unroll
  for (int c = 0; c < 8; ++c) {
    float4 xv = xp[c], sv = sp[c], dv = dp[c];
    ss += xv.x * sv.x + xv.y * sv.y + xv.z * sv.z + xv.w * sv.w;
    ds += xv.x * dv.x + xv.y * dv.y + xv.z * dv.z + xv.w * dv.w;
  }
  a_src[idx] = ss;
  a_dst[idx] = ds;
}

__global__ void gat_fill(float* __restrict__ p, float v, long n) {
  long i = (long)blockIdx.x * blockDim.x + threadIdx.x;
  long stride = (long)gridDim.x * blockDim.x;
  for (; i < n; i += stride) p[i] = v;
}

__device__ __forceinline__ float leaky(float e) {
  return e >= 0.f ? e : NEG_SLOPE * e;
}

// Pass 1: m[dst,h] = max over incoming edges of leaky(a_src[src]+a_dst[dst])
__global__ void __launch_bounds__(256)
gat_edge_max(const int* __restrict__ ei, const float* __restrict__ a_src,
             const float* __restrict__ a_dst, float* __restrict__ m,
             int nE, int nN) {
  int i = blockIdx.x * blockDim.x + threadIdx.x;
  if (i >= nE + nN) return;
  __builtin_prefetch(ei + i + 8192, 0, 0);           // global_prefetch_b8
  int s, d;
  if (i < nE) { s = ei[i]; d = ei[nE + i]; } else { s = d = i - nE; }
  #pragma unroll
  for (int h = 0; h < HEADS; ++h) {
    float e = leaky(a_src[s * HEADS + h] + a_dst[d * HEADS + h]);
    atomicMax(&m[d * HEADS + h], e);                 // global_atomic_max_num_f32
  }
}

// Pass 2: ssum[dst,h] = sum exp(e - m[dst,h])
__global__ void __launch_bounds__(256)
gat_edge_expsum(const int* __restrict__ ei, const float* __restrict__ a_src,
                const float* __restrict__ a_dst, const float* __restrict__ m,
                float* __restrict__ ssum, int nE, int nN) {
  int i = blockIdx.x * blockDim.x + threadIdx.x;
  if (i >= nE + nN) return;
  __builtin_prefetch(ei + i + 8192, 0, 0);
  int s, d;
  if (i < nE) { s = ei[i]; d = ei[nE + i]; } else { s = d = i - nE; }
  #pragma unroll
  for (int h = 0; h < HEADS; ++h) {
    float e = leaky(a_src[s * HEADS + h] + a_dst[d * HEADS + h]);
    atomicAdd(&ssum[d * HEADS + h], __expf(e - m[d * HEADS + h]));
  }
}

// Pass 3: agg[dst,:] += alpha * xh[src,:]. One wave32 per edge; lane L owns
// elements [4L, 4L+4) of the 128-wide feature (head = L>>3), so each edge is
// one coalesced 512B gather + 128 f32 atomic adds spread over 32 lanes.
__global__ void __launch_bounds__(256)
gat_edge_aggregate(const int* __restrict__ ei, const float* __restrict__ a_src,
                   const float* __restrict__ a_dst, const float* __restrict__ m,
                   const float* __restrict__ ssum, const float* __restrict__ xh,
                   float* __restrict__ agg, int nE, int nN) {
  int lane = threadIdx.x & 31;
  int e_id = (int)(((long)blockIdx.x * blockDim.x + threadIdx.x) >> 5);
  if (e_id >= nE + nN) return;
  __builtin_prefetch(ei + e_id + 2048, 0, 0);
  int s, d;
  if (e_id < nE) { s = ei[e_id]; d = ei[nE + e_id]; } else { s = d = e_id - nE; }
  int h = lane >> 3;
  float e = leaky(a_src[s * HEADS + h] + a_dst[d * HEADS + h]);
  float alpha = __expf(e - m[d * HEADS + h]) / ssum[d * HEADS + h];
  int off = lane * 4;
  float4 xv = *(const float4*)(xh + (long)s * FDIM + off);
  float* ap = agg + (long)d * FDIM + off;
  atomicAdd(ap + 0, alpha * xv.x);
  atomicAdd(ap + 1, alpha * xv.y);
  atomicAdd(ap + 2, alpha * xv.z);
  atomicAdd(ap + 3, alpha * xv.w);
}

// h = elu(agg + bias), in place (layer-1 epilogue)
__global__ void __launch_bounds__(256)
gat_bias_elu(float* __restrict__ agg, const float* __restrict__ bias, long total) {
  long i = (long)blockIdx.x * blockDim.x + threadIdx.x;
  if (i >= total) return;
  float v = agg[i] + bias[i & (FDIM - 1)];
  agg[i] = v > 0.f ? v : (__expf(v) - 1.f);
}

// out[n,c] = mean over heads of agg2[n,h,c] + b2[c]  (layer-2 epilogue)
__global__ void __launch_bounds__(256)
gat_head_mean_bias(const float* __restrict__ agg, const float* __restrict__ b2,
                   float* __restrict__ out, int n) {
  int idx = blockIdx.x * blockDim.x + threadIdx.x;   // (node, c) flat
  if (idx >= n * 32) return;
  int node = idx >> 5, c = idx & 31;
  const float* p = agg + (long)node * FDIM + c;
  out[idx] = 0.25f * (p[0] + p[32] + p[64] + p[96]) + b2[c];
}

// ---------------------------------------------------------------------------
extern "C" void kernel_launch(void* const* d_in, const int* in_sizes, int n_in,
                              void* d_out, int out_size, void* d_ws, size_t ws_size,
                              hipStream_t stream) {
  const float* x   = (const float*)d_in[0];
  const int*   ei  = (const int*)  d_in[1];   // [2,E] flat: row0=src, row1=dst
  const float* W1  = (const float*)d_in[2];
  const float* as1 = (const float*)d_in[3];
  const float* ad1 = (const float*)d_in[4];
  const float* b1  = (const float*)d_in[5];
  const float* W2  = (const float*)d_in[6];
  const float* as2 = (const float*)d_in[7];
  const float* ad2 = (const float*)d_in[8];
  const float* b2  = (const float*)d_in[9];
  float* out = (float*)d_out;

  const int N = in_sizes[0] / FDIM;   // 100000
  const int E = in_sizes[1] / 2;      // 1600000
  const int ET = E + N;               // edges incl. self-loops

  // workspace: two N*128 node buffers (ping-pong) + per-(node,head) stats
  float* bufA = (float*)d_ws;                 // xh1, later xh2
  float* bufB = bufA + (long)N * FDIM;        // agg1 -> h1 (in place) -> agg2
  float* aS   = bufB + (long)N * FDIM;
  float* aD   = aS + (long)N * HEADS;
  float* mB   = aD + (long)N * HEADS;
  float* sB   = mB + (long)N * HEADS;

  const int gGemm = N / 16;                          // N divisible by 16
  const int gNH   = (N * HEADS + 255) / 256;
  const int gE    = (ET + 255) / 256;
  const int gAgg  = (int)(((long)ET * 32 + 255) / 256);
  const int gNF   = (int)(((long)N * FDIM + 255) / 256);
  const int gOut  = (N * 32 + 255) / 256;

  // ---------------- Layer 1 ----------------
  gat_gemm128_wmma<<<gGemm, 256, 0, stream>>>(x, W1, bufA);
  gat_att_logits<<<gNH, 256, 0, stream>>>(bufA, as1, ad1, aS, aD, N);
  gat_fill<<<2048, 256, 0, stream>>>(mB, -3.0e38f, (long)N * HEADS);
  gat_fill<<<2048, 256, 0, stream>>>(sB, 0.f, (long)N * HEADS);
  gat_fill<<<4096, 256, 0, stream>>>(bufB, 0.f, (long)N * FDIM);
  gat_edge_max<<<gE, 256, 0, stream>>>(ei, aS, aD, mB, E, N);
  gat_edge_expsum<<<gE, 256, 0, stream>>>(ei, aS, aD, mB, sB, E, N);
  gat_edge_aggregate<<<gAgg, 256, 0, stream>>>(ei, aS, aD, mB, sB, bufA, bufB, E, N);
  gat_bias_elu<<<gNF, 256, 0, stream>>>(bufB, b1, (long)N * FDIM);   // bufB = h1

  // ---------------- Layer 2 ----------------
  gat_gemm128_wmma<<<gGemm, 256, 0, stream>>>(bufB, W2, bufA);       // bufA = xh2
  gat_att_logits<<<gNH, 256, 0, stream>>>(bufA, as2, ad2, aS, aD, N);
  gat_fill<<<2048, 256, 0, stream>>>(mB, -3.0e38f, (long)N * HEADS);
  gat_fill<<<2048, 256, 0, stream>>>(sB, 0.f, (long)N * HEADS);
  gat_fill<<<4096, 256, 0, stream>>>(bufB, 0.f, (long)N * FDIM);     // bufB = agg2
  gat_edge_max<<<gE, 256, 0, stream>>>(ei, aS, aD, mB, E, N);
  gat_edge_expsum<<<gE, 256, 0, stream>>>(ei, aS, aD, mB, sB, E, N);
  gat_edge_aggregate<<<gAgg, 256, 0, stream>>>(ei, aS, aD, mB, sB, bufA, bufB, E, N);
  gat_head_mean_bias<<<gOut, 256, 0, stream>>>(bufB, b2, out, N);
}